// RROIAlign_69544110457142
// MI455X (gfx1250) — compile-verified
//
#include <hip/hip_runtime.h>
#include <cstdint>

#define OUT_HH   7
#define OUT_WW   7
#define NBINS    49
#define NSAMP    196          // 49 bins * 4 samples
#define NCHUNK   13           // ceil(196/16) WMMA chunks
#define NPAD     208          // 13*16, padded so WMMA scatter needs no bounds check
#define FEAT_C   256
#define FEAT_H   200
#define FEAT_W   200
#define SCALE    0.25f

typedef float v2f __attribute__((ext_vector_type(2)));
typedef float v8f __attribute__((ext_vector_type(8)));

__global__ __launch_bounds__(256)
void rroi_align_kernel(const float* __restrict__ feat,
                       const float* __restrict__ rois,
                       float* __restrict__ out)
{
    __shared__ float  s_roi[8];
    __shared__ __align__(16) float s_x[NPAD];
    __shared__ __align__(16) float s_y[NPAD];
    __shared__ int    s_off[NSAMP];
    __shared__ float4 s_w[NSAMP];

    const int n    = blockIdx.x;
    const int tid  = threadIdx.x;
    const int lane = tid & 31;
    const int wave = tid >> 5;

    // ---- Stage the 6 ROI floats into LDS via CDNA5 async-to-LDS ----
    if (tid < 6) {
        uint32_t lds_off = (uint32_t)(uintptr_t)&s_roi[0] + (uint32_t)tid * 4u;
        uint32_t voff    = (uint32_t)tid * 4u;
        uint64_t sbase   = (uint64_t)(uintptr_t)(rois + (size_t)n * 6);
        asm volatile("global_load_async_to_lds_b32 %0, %1, %2"
                     :: "v"(lds_off), "v"(voff), "s"(sbase) : "memory");
    }
    asm volatile("s_wait_asynccnt 0" ::: "memory");
    __syncthreads();

    const int   bidx = (int)s_roi[0];
    const float cx = s_roi[1] * SCALE;
    const float cy = s_roi[2] * SCALE;
    const float rw = fmaxf(s_roi[3] * SCALE, 1.0f);
    const float rh = fmaxf(s_roi[4] * SCALE, 1.0f);
    const float th = s_roi[5];
    const float bw = rw * (1.0f / OUT_WW);
    const float bh = rh * (1.0f / OUT_HH);
    float sinT, cosT;
    sincosf(th, &sinT, &cosT);

    // ---- WMMA affine transform: 16 sample points per v_wmma_f32_16x16x4_f32 ----
    // A (16x4): row m = [xx_m, yy_m, 1, 0]
    //   f32 A layout: lanes 0-15 hold M=lane, K={0,1}; lanes 16-31 hold M=lane-16, K={2,3}
    // B (4x16): cols 0-7 = [cos, sin, cx, 0] (-> x), cols 8-15 = [-sin, cos, cy, 0] (-> y)
    //   B layout (same half-split pattern): VGPR0 = rows K=0 / K=2; VGPR1 = rows K=1 / K=3
    // D (16x16): VGPR j holds M=j (lanes 0-15) and M=j+8 (lanes 16-31), N = lane%16
    for (int chunk = wave; chunk < NCHUNK; chunk += 8) {
        const int m  = lane & 15;
        const int p  = chunk * 16 + m;          // sample point id (196..207 padded)
        const int sx = p & 1;
        const int sy = (p >> 1) & 1;
        const int bin = p >> 2;
        const int pw = bin % OUT_WW;
        const int ph = bin / OUT_WW;
        const float xxv = -0.5f * rw + ((float)pw + ((float)sx + 0.5f) * 0.5f) * bw;
        const float yyv = -0.5f * rh + ((float)ph + ((float)sy + 0.5f) * 0.5f) * bh;

        v2f a, b;
        const bool colx = m < 8;                // columns 0-7 -> x, 8-15 -> y
        if (lane < 16) {                        // K rows {0,1}
            a.x = xxv;  a.y = yyv;
            b.x = colx ? cosT : -sinT;
            b.y = colx ? sinT :  cosT;
        } else {                                // K rows {2,3}
            a.x = 1.0f; a.y = 0.0f;
            b.x = colx ? cx : cy;
            b.y = 0.0f;
        }

        v8f c = {};
        v8f d = __builtin_amdgcn_wmma_f32_16x16x4_f32(
                    false, a, false, b, (short)0, c, false, false);

        if ((lane & 7) == 0) {                  // lanes 0,8,16,24 scatter results
            const bool isx  = ((lane >> 3) & 1) == 0;
            const int  base = chunk * 16 + ((lane >= 16) ? 8 : 0);
            float* dst = isx ? &s_x[base] : &s_y[base];
#pragma unroll
            for (int j = 0; j < 8; ++j)
                dst[j] = d[j];                  // 8 consecutive VGPRs -> contiguous LDS
        }
    }
    __syncthreads();

    // ---- Derive gather offsets + pre-scaled bilinear weights per sample point ----
    if (tid < NSAMP) {
        float x = s_x[tid];
        float y = s_y[tid];
        const bool valid = (y > -1.0f) & (y < (float)FEAT_H) &
                           (x > -1.0f) & (x < (float)FEAT_W);
        y = fminf(fmaxf(y, 0.0f), (float)(FEAT_H - 1));
        x = fminf(fmaxf(x, 0.0f), (float)(FEAT_W - 1));
        const int y0 = (int)fminf(floorf(y), (float)(FEAT_H - 2));
        const int x0 = (int)fminf(floorf(x), (float)(FEAT_W - 2));
        const float ly = y - (float)y0;
        const float lx = x - (float)x0;
        const float hy = 1.0f - ly;
        const float hx = 1.0f - lx;
        const float s = valid ? 0.25f : 0.0f;   // fold validity + 4-sample mean
        s_off[tid] = y0 * FEAT_W + x0;
        s_w[tid]   = make_float4(s * hy * hx, s * hy * lx,
                                 s * ly * hx, s * ly * lx);
    }
    __syncthreads();

    // ---- Main gather loop: thread = channel, loop over 49 bins ----
    const float* base  = feat + ((size_t)bidx * FEAT_C + tid) * (size_t)(FEAT_H * FEAT_W);
    float*       obase = out  + ((size_t)n    * FEAT_C + tid) * (size_t)NBINS;

    for (int bin = 0; bin < NBINS; ++bin) {
        float acc = 0.0f;
#pragma unroll
        for (int s = 0; s < 4; ++s) {
            const int    i = bin * 4 + s;
            const int    o = s_off[i];
            const float4 w = s_w[i];
            acc += w.x * base[o]
                 + w.y * base[o + 1]
                 + w.z * base[o + FEAT_W]
                 + w.w * base[o + FEAT_W + 1];
        }
        obase[bin] = acc;
    }
}

extern "C" void kernel_launch(void* const* d_in, const int* in_sizes, int n_in,
                              void* d_out, int out_size, void* d_ws, size_t ws_size,
                              hipStream_t stream) {
    const float* feat = (const float*)d_in[0];
    const float* rois = (const float*)d_in[1];
    float* out        = (float*)d_out;
    const int N = in_sizes[1] / 6;              // 512 ROIs
    rroi_align_kernel<<<N, 256, 0, stream>>>(feat, rois, out);
    (void)d_ws; (void)ws_size; (void)n_in; (void)out_size;
}